// ShapeContext_82437602279964
// MI455X (gfx1250) — compile-verified
//
#include <hip/hip_runtime.h>

typedef __attribute__((ext_vector_type(2))) float v2f;
typedef __attribute__((ext_vector_type(4))) float v4f;
typedef __attribute__((ext_vector_type(8))) float v8f;
typedef __attribute__((ext_vector_type(4))) int   v4i;

// pointer-to-int4 in global (AS1) and LDS (AS3) address spaces, matching the
// async-to-LDS builtin's parameter types (clang prints AS1 as `__device__`).
typedef v4i __attribute__((address_space(1)))* gbl_v4i_ptr;
typedef v4i __attribute__((address_space(3)))* lds_v4i_ptr;

#define NSITES 131072
#define NIN 32
#define FV 27
#define KDIM 864              // FV * NIN (GEMM reduction dim)
#define NOUT 864
#define ROWS_PER_BLOCK 64     // 4 row-tiles of 16
#define LDS_STRIDE 868        // 864 + 4 pad -> bank-conflict-free A reads

__global__ __launch_bounds__(256)
void shapecontext_wmma(const float* __restrict__ features,
                       const float* __restrict__ weight,
                       const float* __restrict__ bias,
                       const int*   __restrict__ rules,
                       float*       __restrict__ out)
{
    __shared__ float Ald[ROWS_PER_BLOCK * LDS_STRIDE];   // 222,208 B

    const int tid     = threadIdx.x;
    const int rowbase = blockIdx.x * ROWS_PER_BLOCK;

    // ---------------- Stage A (rulebook gather) into LDS ----------------
    // 64 rows x 27 offsets x 8 float4 chunks = 13824 slots
    for (int slot = tid; slot < ROWS_PER_BLOCK * FV * 8; slot += 256) {
        const int part = slot & 7;          // which float4 of the 32-ch row
        const int p    = slot >> 3;
        const int r    = p & (ROWS_PER_BLOCK - 1);
        const int k    = p >> 6;            // filter offset 0..26
        const int rule = rules[k * NSITES + rowbase + r];
        float* dst = &Ald[r * LDS_STRIDE + k * NIN + part * 4];
        if (rule < NSITES) {
            const float* src = features + (size_t)rule * NIN + part * 4;
#if __has_builtin(__builtin_amdgcn_global_load_async_to_lds_b128)
            __builtin_amdgcn_global_load_async_to_lds_b128(
                (gbl_v4i_ptr)src, (lds_v4i_ptr)dst, 0, 0);
#else
            *(v4f*)dst = *(const v4f*)src;
#endif
        } else {
            // sentinel: missing neighbor contributes zeros
            *(v4f*)dst = (v4f){0.f, 0.f, 0.f, 0.f};
        }
    }
#if __has_builtin(__builtin_amdgcn_s_wait_asynccnt)
    __builtin_amdgcn_s_wait_asynccnt(0);
#else
    __asm__ volatile("s_wait_asynccnt 0" ::: "memory");
#endif
    __syncthreads();

    // ---------------- WMMA GEMM: C[64,864] += A[64,864] x B[864,864] ----
    const int w    = tid >> 5;        // wave 0..7
    const int lane = tid & 31;
    const int rt   = w & 3;           // row-tile of this wave
    const int half = lane >> 4;       // 0: K 0/1 (A) | rows 0/1 (B); 1: K 2/3
    const int lm   = lane & 15;       // M index (A) / N index (B,C)

    const float* Arow = &Ald[(rt * 16 + lm) * LDS_STRIDE];

    for (int ct = (w >> 2); ct < NOUT / 16; ct += 2) {
        const int colbase = ct * 16;
        const float* Bcol = weight + colbase + lm;   // column lm of B
        v8f acc = {0.f, 0.f, 0.f, 0.f, 0.f, 0.f, 0.f, 0.f};

        #pragma unroll 8
        for (int kk = 0; kk < KDIM; kk += 4) {
            v2f a, b;
            // A 16x4 fp32 fragment: lanes 0-15 hold K=kk,kk+1; 16-31 K=kk+2,kk+3
            a.x = Arow[kk + 2 * half];
            a.y = Arow[kk + 2 * half + 1];
            // B 4x16 fp32 fragment (mirrored layout), row-major weight
            const float* bp = Bcol + (size_t)(kk + 2 * half) * NOUT;
            b.x = bp[0];
            b.y = bp[NOUT];
            acc = __builtin_amdgcn_wmma_f32_16x16x4_f32(
                      false, a, false, b, (short)0, acc, false, false);
        }

        const float bb = bias[colbase + lm];
        // C/D layout: VGPR j holds M = 8*half + j, N = lm
        float* obase = out + (size_t)(rowbase + rt * 16 + half * 8) * NOUT
                           + colbase + lm;
        #pragma unroll
        for (int j = 0; j < 8; ++j)
            obase[(size_t)j * NOUT] = acc[j] + bb;
    }
}

extern "C" void kernel_launch(void* const* d_in, const int* in_sizes, int n_in,
                              void* d_out, int out_size, void* d_ws, size_t ws_size,
                              hipStream_t stream)
{
    const float* features = (const float*)d_in[0];
    const float* weight   = (const float*)d_in[1];
    const float* bias     = (const float*)d_in[2];
    const int*   rules    = (const int*)d_in[3];
    float*       out      = (float*)d_out;

    dim3 grid(NSITES / ROWS_PER_BLOCK);   // 2048 workgroups
    dim3 block(256);                      // 8 wave32 waves
    hipLaunchKernelGGL(shapecontext_wmma, grid, block, 0, stream,
                       features, weight, bias, rules, out);
}